// JPEGAdversary_1202590843521
// MI455X (gfx1250) — compile-verified
//
#include <hip/hip_runtime.h>
#include <math.h>

typedef __attribute__((ext_vector_type(2))) float v2f;
typedef __attribute__((ext_vector_type(8))) float v8f;

#define WPB 8          // waves per block
#define LSTR 17        // padded LDS row stride (reduces bank conflicts)

__device__ __forceinline__ float dct8(int i, int j) {
    // Orthonormal DCT-II: C8[i][j] = s_i * cos(pi*(2j+1)*i/16)
    float s = (i == 0) ? 0.35355339059327373f : 0.5f;
    return s * __cosf(0.19634954084936207f * (float)((2 * j + 1) * i));
}

__device__ __forceinline__ v8f wmma4(v2f a, v2f b, v8f c) {
    // D = A(16x4) * B(4x16) + C, full fp32
    return __builtin_amdgcn_wmma_f32_16x16x4_f32(
        false, a, false, b, (short)0, c, false, false);
}

__global__ __launch_bounds__(WPB * 32)
void jpeg_adv_kernel(const float* __restrict__ img,
                     const float* __restrict__ noise,
                     float* __restrict__ out)
{
    __shared__ float lds[WPB][16 * LSTR];
    const int wave = threadIdx.x >> 5;
    const int lane = threadIdx.x & 31;
    float* slab = lds[wave];

    const int gtile = blockIdx.x * WPB + wave;   // 0 .. 32767
    const int tx = gtile & 31;                   // tile col (of 32)
    const int ty = (gtile >> 5) & 31;            // tile row
    const int b  = gtile >> 10;                  // batch 0..31

    const int h  = lane >> 4;                    // half-wave select
    const int lm = lane & 15;

    // ---- block-diagonal DCT operands, WMMA A-layout chunks (k = 4c + 2h + r)
    v2f BD[4], BDt[4];
#pragma unroll
    for (int c = 0; c < 4; ++c) {
#pragma unroll
        for (int r = 0; r < 2; ++r) {
            int k = 4 * c + 2 * h + r;
            int m = lm;
            int on = ((m < 8) == (k < 8));
            int i = m & 7, j = k & 7;
            BD[c][r]  = on ? dct8(i, j) : 0.0f;
            BDt[c][r] = on ? dct8(j, i) : 0.0f;
        }
    }

    // ---- load 16x16 RGB tile in WMMA B-layout, convert to YCbCr (centered)
    const size_t chS = 512 * 512;
    const float* imgB = img + (size_t)b * 3 * chS;
    const int row0 = ty * 16, col0 = tx * 16;

    v2f X[3][4];
#pragma unroll
    for (int c = 0; c < 4; ++c) {
#pragma unroll
        for (int r = 0; r < 2; ++r) {
            int row = 4 * c + 2 * h + r;
            size_t off = (size_t)(row0 + row) * 512 + (size_t)(col0 + lm);
            float Rv = imgB[off] * 255.0f;
            float Gv = imgB[chS + off] * 255.0f;
            float Bv = imgB[2 * chS + off] * 255.0f;
            X[0][c][r] = fmaf(0.299f, Rv, fmaf(0.587f, Gv, 0.114f * Bv)) - 128.0f;
            X[1][c][r] = fmaf(-0.168736f, Rv, fmaf(-0.331264f, Gv, 0.5f * Bv));
            X[2][c][r] = fmaf(0.5f, Rv, fmaf(-0.418688f, Gv, -0.081312f * Bv));
        }
    }

    // noise addressing: comp^T(m=r+8h, n=lm) == comp(8a+u, 8h+v), a=lm>>3,u=lm&7,v=r
    const int u = lm & 7, a = lm >> 3;
    const size_t pbase =
        ((size_t)((2 * ty + a) * 64 + (2 * tx + h))) * 64 + (size_t)u * 8;

    v8f rec[3];
#pragma unroll
    for (int ch = 0; ch < 3; ++ch) {
        // Stage 1: P = BD @ X           (D layout)
        v8f acc1 = {};
#pragma unroll
        for (int c = 0; c < 4; ++c) acc1 = wmma4(BD[c], X[ch][c], acc1);

        // LDS transpose: P -> P^T in B-layout
#pragma unroll
        for (int r = 0; r < 8; ++r) slab[(r + 8 * h) * LSTR + lm] = acc1[r];
        v2f Bop[4];
#pragma unroll
        for (int c = 0; c < 4; ++c)
#pragma unroll
            for (int r = 0; r < 2; ++r)
                Bop[c][r] = slab[lm * LSTR + (4 * c + 2 * h + r)];

        // Stage 2: comp^T = BD @ P^T = BD X^T BDt
        v8f acc2 = {};
#pragma unroll
        for (int c = 0; c < 4; ++c) acc2 = wmma4(BD[c], Bop[c], acc2);

        // adversarial quantization noise
        const float* np = noise + ((size_t)b * 3 + ch) * (4096u * 64u) + pbase;
#pragma unroll
        for (int r = 0; r < 8; ++r) acc2[r] = fmaf(255.0f, np[r], acc2[r]);

        // LDS pass-through: comp^T -> B-layout (no transpose)
#pragma unroll
        for (int r = 0; r < 8; ++r) slab[(r + 8 * h) * LSTR + lm] = acc2[r];
#pragma unroll
        for (int c = 0; c < 4; ++c)
#pragma unroll
            for (int r = 0; r < 2; ++r)
                Bop[c][r] = slab[(4 * c + 2 * h + r) * LSTR + lm];

        // Stage 3: Y1 = BDt @ comp^T
        v8f acc3 = {};
#pragma unroll
        for (int c = 0; c < 4; ++c) acc3 = wmma4(BDt[c], Bop[c], acc3);

        // LDS transpose: Y1 -> Y1^T = comp @ BD in B-layout
#pragma unroll
        for (int r = 0; r < 8; ++r) slab[(r + 8 * h) * LSTR + lm] = acc3[r];
#pragma unroll
        for (int c = 0; c < 4; ++c)
#pragma unroll
            for (int r = 0; r < 2; ++r)
                Bop[c][r] = slab[lm * LSTR + (4 * c + 2 * h + r)];

        // Stage 4: rec = BDt @ comp @ BD   (D layout, spatial orientation)
        v8f acc4 = {};
#pragma unroll
        for (int c = 0; c < 4; ++c) acc4 = wmma4(BDt[c], Bop[c], acc4);
        rec[ch] = acc4;
    }

    // ---- YCbCr -> RGB, clip, store (D layout rows are contiguous)
    float* outB = out + (size_t)b * 3 * chS;
#pragma unroll
    for (int r = 0; r < 8; ++r) {
        float Yv  = rec[0][r] + 128.0f;   // +128 from IDCT shift
        float Cbv = rec[1][r];            // (+128 then -128) cancels
        float Crv = rec[2][r];
        float Rv = fmaf(1.402f, Crv, Yv);
        float Gv = fmaf(-0.714136f, Crv, fmaf(-0.344136f, Cbv, Yv));
        float Bv = fmaf(1.772f, Cbv, Yv);
        size_t off = (size_t)(row0 + r + 8 * h) * 512 + (size_t)(col0 + lm);
        outB[off]           = fminf(fmaxf(Rv * (1.0f / 255.0f), 0.0f), 1.0f);
        outB[chS + off]     = fminf(fmaxf(Gv * (1.0f / 255.0f), 0.0f), 1.0f);
        outB[2 * chS + off] = fminf(fmaxf(Bv * (1.0f / 255.0f), 0.0f), 1.0f);
    }
}

extern "C" void kernel_launch(void* const* d_in, const int* in_sizes, int n_in,
                              void* d_out, int out_size, void* d_ws, size_t ws_size,
                              hipStream_t stream) {
    const float* img   = (const float*)d_in[0];   // (32,3,512,512) f32
    const float* noise = (const float*)d_in[1];   // (32,3,4096,8,8) f32
    float* out = (float*)d_out;                   // (32,3,512,512) f32

    // 32 batches * 32*32 tiles = 32768 waves; 8 waves per 256-thread block
    dim3 grid(32768 / WPB), block(WPB * 32);
    jpeg_adv_kernel<<<grid, block, 0, stream>>>(img, noise, out);
}